// CBOW_88991722373900
// MI455X (gfx1250) — compile-verified
//
#include <hip/hip_runtime.h>
#include <math.h>

// ---------------------------------------------------------------------------
// CBOW forward for MI455X (gfx1250, wave32).
// Roofline: 312 MB streamed / 23.3 TB/s ~= 13.4 us; 154 MFLOP -> pure BW bound.
// Strategy: fp32 end-to-end, stream W_in / W_out exactly once with b64 loads,
// do all MACs with V_WMMA_F32_16X16X4_F32 (A = 16-row weight tile, B = vector
// replicated across the 16 columns; columns of D are identical by design).
// ---------------------------------------------------------------------------

typedef float v2f __attribute__((ext_vector_type(2)));
typedef float v4f __attribute__((ext_vector_type(4)));
typedef float v8f __attribute__((ext_vector_type(8)));

#define V_DIM   128000
#define N_DIM   300
#define C_DIM   10
#define KCHUNKS 128
#define VCHUNK  (V_DIM / KCHUNKS)   // 1000
#define NTILES  19                  // ceil(300/16)
#define NPAD    304
#define V2TILES (V_DIM / 16)        // 8000
#define K2BLOCKS (V2TILES / 8)      // 1000 blocks of 8 waves
#define SMBLOCKS 125                // 125 * 1024 == 128000

// Monotonic unsigned key for order-independent float max via atomicMax.
__device__ __forceinline__ unsigned fkey(float f) {
  unsigned u = __float_as_uint(f);
  return (u & 0x80000000u) ? ~u : (u | 0x80000000u);
}
__device__ __forceinline__ float funkey(unsigned k) {
  unsigned u = (k & 0x80000000u) ? (k & 0x7fffffffu) : ~k;
  return __uint_as_float(u);
}

// k0: s[v] = (1/C) * sum_c ctx[c,v]   (5.12 MB read, trivially coalesced)
__global__ __launch_bounds__(256) void ctx_reduce(const float* __restrict__ ctx,
                                                  float* __restrict__ s) {
  int v = blockIdx.x * 256 + threadIdx.x;           // grid = V/256 exact
  float acc = 0.0f;
#pragma unroll
  for (int c = 0; c < C_DIM; ++c) acc += ctx[(size_t)c * V_DIM + v];
  s[v] = acc * (1.0f / (float)C_DIM);
}

// k1: hidden partials via WMMA.  One wave = one (n-tile, V-chunk).
// A[m][k] = W_in[n0+m][v+k], B[k][j] = s[v+k] (replicated over columns).
__global__ __launch_bounds__(32) void gemv1_wmma(const float* __restrict__ Win,
                                                 const float* __restrict__ s,
                                                 float* __restrict__ hpart) {
  const int lane  = threadIdx.x;        // wave32, one wave per block
  const int tile  = blockIdx.x;         // 0..18  (n tile)
  const int chunk = blockIdx.y;         // 0..127 (V chunk)
  const int half  = lane >> 4;          // 0: K=0,1   1: K=2,3
  const int r     = lane & 15;
  int row = tile * 16 + r;
  if (row > N_DIM - 1) row = N_DIM - 1; // clamp padded rows (results discarded)
  const float* arow = Win + (size_t)row * V_DIM + half * 2;
  const float* brow = s + half * 2;
  const int v0 = chunk * VCHUNK;

  v8f acc = {};
  for (int v = v0; v < v0 + VCHUNK; v += 8) {       // 125 iters, 2 WMMA each
    v2f a0 = *(const v2f*)(arow + v);
    v2f b0 = *(const v2f*)(brow + v);
    v2f a1 = *(const v2f*)(arow + v + 4);
    v2f b1 = *(const v2f*)(brow + v + 4);
    acc = __builtin_amdgcn_wmma_f32_16x16x4_f32(false, a0, false, b0,
                                                (short)0, acc, false, false);
    acc = __builtin_amdgcn_wmma_f32_16x16x4_f32(false, a1, false, b1,
                                                (short)0, acc, false, false);
  }

  // D columns identical: lane 0 holds rows 0..7, lane 16 holds rows 8..15.
  if (r == 0) {
    union { v8f v; float f[8]; } u; u.v = acc;
    const int base = tile * 16 + half * 8;
#pragma unroll
    for (int i = 0; i < 8; ++i) {
      int n = base + i;
      if (n < N_DIM) hpart[chunk * NPAD + n] = u.f[i];   // race-free slot
    }
  }
}

// k1b: deterministic reduction of 128 partials -> hidden[0..303]
__global__ __launch_bounds__(320) void hidden_reduce(const float* __restrict__ hpart,
                                                     float* __restrict__ hidden) {
  int n = threadIdx.x;
  if (n < NPAD) {
    float a = 0.0f;
    if (n < N_DIM)
      for (int c = 0; c < KCHUNKS; ++c) a += hpart[c * NPAD + n];
    hidden[n] = (n < N_DIM) ? a : 0.0f;
  }
}

// k2: logits via WMMA + per-block max.  8 waves/block, wave = one 16-row v tile.
// A[m][k] = W_out[v0+m][n+k], B from LDS-staged hidden.
__global__ __launch_bounds__(256) void gemv2_wmma(const float* __restrict__ Wout,
                                                  const float* __restrict__ hidden,
                                                  float* __restrict__ logits,
                                                  unsigned* __restrict__ bmax) {
  __shared__ __align__(16) float h[NPAD];
  __shared__ unsigned smax;
  const int t = threadIdx.x;
  if (t == 0) smax = 0u;                              // min key
  for (int i = t; i < NPAD; i += 256) h[i] = (i < N_DIM) ? hidden[i] : 0.0f;
  __syncthreads();

  const int wave = t >> 5, lane = t & 31;
  const int half = lane >> 4, r = lane & 15;
  const int tile = blockIdx.x * 8 + wave;             // 0..7999
  const int v0   = tile * 16;
  const float* arow = Wout + (size_t)(v0 + r) * N_DIM + half * 2;
  const float* brow = h + half * 2;

  v8f acc = {};
  for (int n = 0; n < N_DIM; n += 4) {                // 75 WMMA / wave
    v2f a = *(const v2f*)(arow + n);                  // global_load_b64
    v2f b = *(const v2f*)(brow + n);                  // ds_load_b64
    acc = __builtin_amdgcn_wmma_f32_16x16x4_f32(false, a, false, b,
                                                (short)0, acc, false, false);
  }

  unsigned lkey = 0u;
  if (r == 0) {
    union { v8f v; float f[8]; } u; u.v = acc;
    const int base = v0 + half * 8;
    v4f lo = { u.f[0], u.f[1], u.f[2], u.f[3] };
    v4f hi = { u.f[4], u.f[5], u.f[6], u.f[7] };
    *(v4f*)(logits + base)     = lo;                  // 16B-aligned b128 stores
    *(v4f*)(logits + base + 4) = hi;
    float m = u.f[0];
#pragma unroll
    for (int i = 1; i < 8; ++i) m = fmaxf(m, u.f[i]);
    lkey = fkey(m);
  }
  atomicMax(&smax, lkey);                             // order-independent
  __syncthreads();
  if (t == 0) bmax[blockIdx.x] = smax;
}

// k3: per-block exp-sums (deterministic tree reduction)
__global__ __launch_bounds__(1024) void softmax_sum(const float* __restrict__ logits,
                                                    const unsigned* __restrict__ bmax,
                                                    float* __restrict__ bsum) {
  __shared__ float ssum[1024];
  __shared__ float sgmax;
  const int t = threadIdx.x;
  if (t == 0) {
    unsigned k = 0u;
    for (int i = 0; i < K2BLOCKS; ++i) k = max(k, bmax[i]);
    sgmax = funkey(k);
  }
  __syncthreads();
  const float gmax = sgmax;
  const int v = blockIdx.x * 1024 + t;                // exact coverage of V
  ssum[t] = expf(logits[v] - gmax);
  __syncthreads();
  for (int off = 512; off > 0; off >>= 1) {
    if (t < off) ssum[t] += ssum[t + off];
    __syncthreads();
  }
  if (t == 0) bsum[blockIdx.x] = ssum[0];
}

// k4: normalize in place (d_out holds logits, becomes softmax)
__global__ __launch_bounds__(1024) void softmax_norm(float* __restrict__ logits,
                                                     const unsigned* __restrict__ bmax,
                                                     const float* __restrict__ bsum) {
  __shared__ float sgmax, sinv;
  const int t = threadIdx.x;
  if (t == 0) {
    unsigned k = 0u;
    for (int i = 0; i < K2BLOCKS; ++i) k = max(k, bmax[i]);
    sgmax = funkey(k);
    float g = 0.0f;
    for (int i = 0; i < SMBLOCKS; ++i) g += bsum[i];
    sinv = 1.0f / g;
  }
  __syncthreads();
  const int v = blockIdx.x * 1024 + t;
  logits[v] = expf(logits[v] - sgmax) * sinv;
}

extern "C" void kernel_launch(void* const* d_in, const int* in_sizes, int n_in,
                              void* d_out, int out_size, void* d_ws, size_t ws_size,
                              hipStream_t stream) {
  const float* ctx  = (const float*)d_in[0];   // [C, V]
  const float* Win  = (const float*)d_in[1];   // [N, V]
  const float* Wout = (const float*)d_in[2];   // [V, N]
  float* out = (float*)d_out;                  // [V] logits -> softmax in place

  // Workspace layout (~673 KB): s | hpart | hidden | bmax | bsum
  float*    s      = (float*)d_ws;
  float*    hpart  = s + V_DIM;
  float*    hidden = hpart + (size_t)KCHUNKS * NPAD;
  unsigned* bmax   = (unsigned*)(hidden + NPAD);
  float*    bsum   = (float*)(bmax + K2BLOCKS);

  ctx_reduce   <<<V_DIM / 256, 256, 0, stream>>>(ctx, s);
  gemv1_wmma   <<<dim3(NTILES, KCHUNKS), 32, 0, stream>>>(Win, s, hpart);
  hidden_reduce<<<1, 320, 0, stream>>>(hpart, hidden);
  gemv2_wmma   <<<K2BLOCKS, 256, 0, stream>>>(Wout, hidden, out, bmax);
  softmax_sum  <<<SMBLOCKS, 1024, 0, stream>>>(out, bmax, bsum);
  softmax_norm <<<SMBLOCKS, 1024, 0, stream>>>(out, bmax, bsum);
}